// Model_inGeo_14946486190731
// MI455X (gfx1250) — compile-verified
//
#include <hip/hip_runtime.h>

#define NN 100000
#define NE 1600000
#define D 64
#define LDSP 66   // LDS pitch (floats) to break 256B bank-wrap conflicts

typedef __attribute__((ext_vector_type(2))) float v2f;
typedef __attribute__((ext_vector_type(8))) float v8f;

// ---------------------------------------------------------------- fill zeros
__global__ void fill_zero_kernel(float4* __restrict__ p, int n4) {
    int i = blockIdx.x * blockDim.x + threadIdx.x;
    if (i < n4) p[i] = make_float4(0.f, 0.f, 0.f, 0.f);
}

// ------------------------------------------------- edge scatter (segment max)
// h >= 0 (relu output) so u32 bit-compare == f32 compare; agg pre-filled 0.
__global__ void edge_scatter_max_kernel(const int* __restrict__ src,
                                        const int* __restrict__ dst,
                                        const float* __restrict__ h,
                                        unsigned int* __restrict__ agg,
                                        int nedges) {
    long long t = (long long)blockIdx.x * blockDim.x + threadIdx.x;
    int e = (int)(t >> 4);
    if (e >= nedges) return;
    int f = (int)(t & 15) << 2;
    int s = src[e];
    int d = dst[e];
    const float4 v = *(const float4*)(h + (size_t)s * D + f);
    unsigned int* p = agg + (size_t)d * D + f;
    atomicMax(p + 0, __float_as_uint(v.x));
    atomicMax(p + 1, __float_as_uint(v.y));
    atomicMax(p + 2, __float_as_uint(v.z));
    atomicMax(p + 3, __float_as_uint(v.w));
}

// ------------------------------------------------------------- WMMA GEMM
// out[n, o] = act( sum_k A1[n,k]*W1[o,k] (+ sum_k A2[n,k]*W2[o,k]) + bias[o] )
// One wave handles 16 rows x 64 cols using V_WMMA_F32_16X16X4_F32.
template <bool RELU, bool DUAL>
__global__ __launch_bounds__(256) void sage_gemm_kernel(
    const float* __restrict__ A1, const float* __restrict__ W1,
    const float* __restrict__ A2, const float* __restrict__ W2,
    const float* __restrict__ bias, float* __restrict__ out, int nrows) {
    __shared__ float ldsW1[D * LDSP];
    __shared__ float ldsW2[DUAL ? (D * LDSP) : 1];

    int tid = threadIdx.x;
    for (int i = tid; i < D * D; i += 256) {
        int r = i >> 6, c = i & 63;
        ldsW1[r * LDSP + c] = W1[i];
        if (DUAL) ldsW2[r * LDSP + c] = W2[i];
    }
    __syncthreads();

    int lane = tid & 31;
    int wave = tid >> 5;
    int tile = blockIdx.x * 8 + wave;     // wave-uniform
    int row0 = tile * 16;
    if (row0 >= nrows) return;            // uniform branch: EXEC stays all-ones

    int m = lane & 15;                    // row within tile (A) / col within tile (B)
    int khalf = (lane >> 4) << 1;         // 0 for lanes 0-15, 2 for lanes 16-31
    int rowA = row0 + m;
    if (rowA >= nrows) rowA = nrows - 1;  // clamp; masked at store
    const float* a1base = A1 + (size_t)rowA * D;
    const float* a2base = DUAL ? (A2 + (size_t)rowA * D) : nullptr;

    v8f acc[4];
#pragma unroll
    for (int ct = 0; ct < 4; ++ct) {
        float bv = bias[ct * 16 + m];
        acc[ct] = (v8f){bv, bv, bv, bv, bv, bv, bv, bv};
    }

#pragma unroll
    for (int kb = 0; kb < D; kb += 4) {
        v2f a = *(const v2f*)(a1base + kb + khalf);
#pragma unroll
        for (int ct = 0; ct < 4; ++ct) {
            v2f b = *(const v2f*)(&ldsW1[(ct * 16 + m) * LDSP + kb + khalf]);
            acc[ct] = __builtin_amdgcn_wmma_f32_16x16x4_f32(
                false, a, false, b, (short)0, acc[ct], false, false);
        }
    }
    if (DUAL) {
#pragma unroll
        for (int kb = 0; kb < D; kb += 4) {
            v2f a = *(const v2f*)(a2base + kb + khalf);
#pragma unroll
            for (int ct = 0; ct < 4; ++ct) {
                v2f b = *(const v2f*)(&ldsW2[(ct * 16 + m) * LDSP + kb + khalf]);
                acc[ct] = __builtin_amdgcn_wmma_f32_16x16x4_f32(
                    false, a, false, b, (short)0, acc[ct], false, false);
            }
        }
    }

    // C/D layout: VGPR r -> M = r + 8*(lane>=16); N = lane & 15
    int rbase = (lane >> 4) << 3;
#pragma unroll
    for (int r = 0; r < 8; ++r) {
        int row = row0 + rbase + r;
        if (row < nrows) {
#pragma unroll
            for (int ct = 0; ct < 4; ++ct) {
                float v = acc[ct][r];
                if (RELU) v = fmaxf(v, 0.f);
                out[(size_t)row * D + ct * 16 + m] = v;
            }
        }
    }
}

// ---------------------------------------------------------------- launcher
extern "C" void kernel_launch(void* const* d_in, const int* in_sizes, int n_in,
                              void* d_out, int out_size, void* d_ws, size_t ws_size,
                              hipStream_t stream) {
    const float* x   = (const float*)d_in[0];
    const int*   src = (const int*)d_in[1];
    const int*   dst = (const int*)d_in[2];
    const float* Wp1 = (const float*)d_in[3];
    const float* bp1 = (const float*)d_in[4];
    const float* Ws1 = (const float*)d_in[5];
    const float* Wn1 = (const float*)d_in[6];
    const float* b1  = (const float*)d_in[7];
    const float* Wp2 = (const float*)d_in[8];
    const float* bp2 = (const float*)d_in[9];
    const float* Ws2 = (const float*)d_in[10];
    const float* Wn2 = (const float*)d_in[11];
    const float* b2  = (const float*)d_in[12];

    float* ws     = (float*)d_ws;
    float* h_pool = ws;
    float* agg    = ws + (size_t)NN * D;
    float* h1     = ws + 2 * (size_t)NN * D;
    float* outp   = (float*)d_out;

    const int nTiles     = (NN + 15) / 16;
    const int gemmBlocks = (nTiles + 7) / 8;
    const int fillN4     = NN * D / 4;
    const int fillBlocks = (fillN4 + 255) / 256;
    const long long eThreads = (long long)NE * 16;
    const int eBlocks    = (int)((eThreads + 255) / 256);

    // ---- layer 1
    sage_gemm_kernel<true, false><<<gemmBlocks, 256, 0, stream>>>(
        x, Wp1, nullptr, nullptr, bp1, h_pool, NN);
    fill_zero_kernel<<<fillBlocks, 256, 0, stream>>>((float4*)agg, fillN4);
    edge_scatter_max_kernel<<<eBlocks, 256, 0, stream>>>(
        src, dst, h_pool, (unsigned int*)agg, NE);
    sage_gemm_kernel<false, true><<<gemmBlocks, 256, 0, stream>>>(
        x, Ws1, agg, Wn1, b1, h1, NN);

    // ---- layer 2
    sage_gemm_kernel<true, false><<<gemmBlocks, 256, 0, stream>>>(
        h1, Wp2, nullptr, nullptr, bp2, h_pool, NN);
    fill_zero_kernel<<<fillBlocks, 256, 0, stream>>>((float4*)agg, fillN4);
    edge_scatter_max_kernel<<<eBlocks, 256, 0, stream>>>(
        src, dst, h_pool, (unsigned int*)agg, NE);
    sage_gemm_kernel<false, true><<<gemmBlocks, 256, 0, stream>>>(
        h1, Ws2, agg, Wn2, b2, outp, NN);
}